// W_BCE_IOU_loss_12111807774996
// MI455X (gfx1250) — compile-verified
//
#include <hip/hip_runtime.h>
#include <math.h>

// MI455X / gfx1250, wave32. Fused weighted-BCE+IoU loss.
// Box filter: vertical = per-lane running window sums (2 adds/elt),
// horizontal = banded-matrix GEMM on the WMMA pipe (V_WMMA_F32_16X16X4_F32).

typedef float v2f __attribute__((ext_vector_type(2)));
typedef float v8f __attribute__((ext_vector_type(8)));

#define BATCH 64
#define H     512
#define W     512
#define HW    (H * W)
#define KW    31
#define RAD   15
#define SW    128              // output columns per block
#define LCOLS (SW + 2 * RAD)   // 158 real columns (incl. halo)
#define VW    161              // padded LDS row width (odd stride -> no bank conflicts)
#define NSTRIP (W / SW)

__global__ __launch_bounds__(256)
void wbce_iou_main(const float* __restrict__ pred,
                   const float* __restrict__ mask,
                   float* __restrict__ ws)
{
    __shared__ float ring[32][VW];   // last 32 mask rows of this strip (+halo)
    __shared__ float vbuf[16][VW];   // 16 rows of vertical 31-sums
    __shared__ float red[256];

    const int t  = threadIdx.x;
    const int b  = blockIdx.y;
    const int x0 = blockIdx.x * SW;

    // ---- phase-A column ownership (one lane per column, incl. halo + pad) ----
    const int  gx     = x0 - RAD + t;
    const bool colAct = (t < VW);
    const bool inCol  = colAct && (gx >= 0) && (gx < W);
    const float* mbase = mask + (size_t)b * HW;
    const float* pbase = pred + (size_t)b * HW;

    // ---- WMMA operand lane decomposition (16x16x4 f32 layouts) ----
    const int lane = t & 31;
    const int wv   = t >> 5;      // 8 waves -> 8 column tiles of 16
    const int lo   = lane & 15;   // A: M row / B,D: N column
    const int hi   = lane >> 4;   // A,B: K pair select / D: M half select

    // ---- vertical running-sum prologue: mask rows 0..RAD-1 ----
    float vs = 0.0f;
    for (int yy = 0; yy < RAD; ++yy) {
        float val = inCol ? mbase[yy * W + gx] : 0.0f;
        if (colAct) ring[yy & 31][t] = val;
        vs += val;
    }

    float t_bce = 0.0f, t_inter = 0.0f, t_union = 0.0f;
    const float inv_area = 1.0f / (float)(KW * KW);

    for (int Y = 0; Y < H; Y += 16) {
        // prefetch this group's pred rows (L2-resident; emits global_prefetch_b8)
        {
            const int prow = Y + (t >> 4);
            const int pcol = x0 + (t & 15) * 8;
            __builtin_prefetch(pbase + prow * W + pcol, 0, 1);
        }

        // ---- phase A: advance vertical window 16 rows ----
        for (int r = 0; r < 16; ++r) {
            const int y = Y + r;
            if (colAct) {
                const int ynew = y + RAD;
                float val = (inCol && ynew < H) ? mbase[ynew * W + gx] : 0.0f;
                ring[ynew & 31][t] = val;
                vs += val;
                const int yold = y - RAD - 1;
                if (yold >= 0) vs -= ring[yold & 31][t];
                vbuf[r][t] = vs;
            }
        }
        __syncthreads();

        // ---- phase B: horizontal 31-tap = V(16x48) x Band(48x16) on WMMA pipe ----
        // Band[k][n] = 1 iff 0 <= k - n <= 30  (zero-padded K to 48).
        const int c0 = 16 * wv;   // local column base of this tile's A panel
        v8f acc = {0.f, 0.f, 0.f, 0.f, 0.f, 0.f, 0.f, 0.f};
#pragma unroll
        for (int kk = 0; kk < 12; ++kk) {
            const int kb = 4 * kk;
            v2f A, Bm;
            // A 16x4 f32: lanes 0-15 hold K={0,1}, lanes 16-31 hold K={2,3}; M = lane&15
            A.x = vbuf[lo][c0 + kb + 2 * hi];
            A.y = vbuf[lo][c0 + kb + 2 * hi + 1];
            // B 4x16 f32 (mirror layout): N = lane&15, K = {2*hi, 2*hi+1}
            const int d0 = kb + 2 * hi - lo;
            Bm.x = (d0 >= 0     && d0     <= 2 * RAD) ? 1.0f : 0.0f;
            Bm.y = (d0 + 1 >= 0 && d0 + 1 <= 2 * RAD) ? 1.0f : 0.0f;
            acc = __builtin_amdgcn_wmma_f32_16x16x4_f32(
                false, A, false, Bm, (short)0, acc, false, false);
        }

        // ---- phase C: elementwise loss + local reduction ----
        // D layout: element (M = vr + 8*hi, N = lo) in acc[vr]
        const int ocol = x0 + 16 * wv + lo;      // global output column
        const int lctr = RAD + 16 * wv + lo;     // ring column of the center mask
#pragma unroll
        for (int vr = 0; vr < 8; ++vr) {
            const int row = vr + 8 * hi;
            const int y   = Y + row;
            const float m = ring[y & 31][lctr];
            const float z = pbase[y * W + ocol];
            const float avg  = acc[vr] * inv_area;
            const float weit = 1.0f + 5.0f * fabsf(avg - m);
            const float p    = 1.0f / (1.0f + __expf(-z));
            const float bce  = fmaxf(z, 0.0f) - z * m + log1pf(__expf(-fabsf(z)));
            t_bce   += bce;
            t_inter += p * m * weit;
            t_union += (p + m) * weit;
        }
        __syncthreads();   // protect ring/vbuf before next group's phase A
    }

    // ---- block reduction of the three partials ----
    float blk[3];
    float vals[3] = {t_bce, t_inter, t_union};
#pragma unroll
    for (int q = 0; q < 3; ++q) {
        red[t] = vals[q];
        __syncthreads();
        for (int s = 128; s > 0; s >>= 1) {
            if (t < s) red[t] += red[t + s];
            __syncthreads();
        }
        blk[q] = red[0];
        __syncthreads();
    }
    if (t == 0) {
        atomicAdd(&ws[0],             blk[0]);
        atomicAdd(&ws[1 + b],         blk[1]);
        atomicAdd(&ws[1 + BATCH + b], blk[2]);
    }
}

__global__ void wbce_zero(float* ws)
{
    if (threadIdx.x < 1 + 2 * BATCH) ws[threadIdx.x] = 0.0f;
}

__global__ void wbce_final(const float* __restrict__ ws, float* __restrict__ out)
{
    __shared__ float sh[BATCH];
    const int t = threadIdx.x;
    const float inter = ws[1 + t];
    const float uni   = ws[1 + BATCH + t];
    sh[t] = 1.0f - (inter + 1.0f) / (uni - inter + 1.0f);   // wiou_b
    __syncthreads();
    for (int s = BATCH / 2; s > 0; s >>= 1) {
        if (t < s) sh[t] += sh[t + s];
        __syncthreads();
    }
    if (t == 0) {
        const float wbce = ws[0] / (float)((size_t)BATCH * HW);
        out[0] = wbce + sh[0] / (float)BATCH;
    }
}

extern "C" void kernel_launch(void* const* d_in, const int* in_sizes, int n_in,
                              void* d_out, int out_size, void* d_ws, size_t ws_size,
                              hipStream_t stream)
{
    const float* pred = (const float*)d_in[0];
    const float* mask = (const float*)d_in[1];
    float* out = (float*)d_out;
    float* ws  = (float*)d_ws;

    hipLaunchKernelGGL(wbce_zero,    dim3(1),             dim3(256), 0, stream, ws);
    hipLaunchKernelGGL(wbce_iou_main, dim3(NSTRIP, BATCH), dim3(256), 0, stream,
                       pred, mask, ws);
    hipLaunchKernelGGL(wbce_final,   dim3(1),             dim3(BATCH), 0, stream, ws, out);
}